// SiameseNet_24068996726829
// MI455X (gfx1250) — compile-verified
//
#include <hip/hip_runtime.h>
#include <hip/hip_bf16.h>

typedef __attribute__((ext_vector_type(2))) float v2f;
typedef __attribute__((ext_vector_type(8))) float v8f;

#define BN_N   16
#define BN_C   256
#define BN_K   6
#define BN_M   22
#define BN_HO  17               // 22 - 6 + 1
#define BN_NP  (BN_HO * BN_HO)  // 289
#define BN_UV  (BN_K * BN_K)    // 36
#define SX_ELEMS  (BN_N * BN_C * BN_M * BN_M)   // 1,982,464
#define SZW_ELEMS (BN_N * BN_C * BN_UV)         // 147,456
#define NTILES 19               // ceil(289/16)

// ---------------- Kernel 1: sqrt(x), w_c*sqrt(z) ----------------
__global__ void siam_prep_kernel(const float* __restrict__ z,
                                 const float* __restrict__ x,
                                 const float* __restrict__ w,
                                 float* __restrict__ sx,
                                 float* __restrict__ szw) {
    int total = SX_ELEMS + SZW_ELEMS;
    for (int i = blockIdx.x * blockDim.x + threadIdx.x; i < total;
         i += gridDim.x * blockDim.x) {
        if (i < SX_ELEMS) {
            sx[i] = sqrtf(x[i]);
        } else {
            int j = i - SX_ELEMS;
            int c = (j / BN_UV) % BN_C;     // layout [n, c, u, v]
            szw[j] = sqrtf(z[j]) * w[c];
        }
    }
}

// ---------------- Kernel 2: im2col-GEMM via V_WMMA_F32_16X16X4_F32 ----------
// One wave32 per (n, tile of 16 output positions). D = A(16x4) * B(4x16) + C,
// reduction K = 9216 walked 4 at a time; B column-replicated so col 0 holds
// the GEMV result. A lane layout (ISA 7.12.2, 32-bit A 16x4):
//   lanes 0-15:  VGPR0 = K0, VGPR1 = K1   (M = lane)
//   lanes 16-31: VGPR0 = K2, VGPR1 = K3   (M = lane-16)
// B mirrors that (4x16 rows striped: v0 = {K0|K2}, v1 = {K1|K3}).
__global__ __launch_bounds__(32) void siam_corr_kernel(
        const float* __restrict__ sx,
        const float* __restrict__ szw,
        float* __restrict__ meanc) {
    __shared__ float sB[BN_C * BN_UV];       // 36 KB: w*sqrt(z) for this n

    const int bid  = blockIdx.x;             // 0..303
    const int n    = bid / NTILES;
    const int tile = bid % NTILES;
    const int lane = threadIdx.x;            // wave32
    const int row  = lane & 15;
    const int kx   = (lane >> 4) << 1;       // 0 (lanes 0-15) or 2 (lanes 16-31)

    const float* __restrict__ bzn = szw + (size_t)n * (BN_C * BN_UV);
    for (int i = lane; i < BN_C * BN_UV; i += 32) sB[i] = bzn[i];
    __syncthreads();

    int p  = tile * 16 + row;
    int pe = p < BN_NP ? p : BN_NP - 1;      // clamp pad rows (branchless)
    int i0 = pe / BN_HO;
    int j0 = pe - i0 * BN_HO;

    const float* __restrict__ sxn = sx + (size_t)n * (BN_C * BN_M * BN_M);

    v8f acc = {};
    for (int uv0 = 0; uv0 < BN_UV; uv0 += 4) {
        const int uvx = uv0 + kx;            // this lane's K slots (VGPR0, VGPR1)
        const int uvy = uvx + 1;
        const int ux = uvx / BN_K, vx = uvx - ux * BN_K;
        const int uy = uvy / BN_K, vy = uvy - uy * BN_K;
        const float* __restrict__ pax = sxn + (i0 + ux) * BN_M + (j0 + vx);
        const float* __restrict__ pay = sxn + (i0 + uy) * BN_M + (j0 + vy);
        const float* __restrict__ pbx = sB + uvx;
        const float* __restrict__ pby = sB + uvy;
#pragma unroll 4
        for (int c = 0; c < BN_C; ++c) {
            v2f a, b;
            a.x = pax[c * (BN_M * BN_M)];
            a.y = pay[c * (BN_M * BN_M)];
            b.x = pbx[c * BN_UV];
            b.y = pby[c * BN_UV];
            acc = __builtin_amdgcn_wmma_f32_16x16x4_f32(
                false, a, false, b, (short)0, acc, false, false);
        }
    }

    // Column 0 of D lives in lane 0 (rows 0-7) and lane 16 (rows 8-15).
    if ((lane & 15) == 0) {
        const float inv36 = 1.0f / 36.0f;
        int rbase = tile * 16 + ((lane >> 4) << 3);
#pragma unroll
        for (int r = 0; r < 8; ++r) {
            int pp = rbase + r;
            if (pp < BN_NP) meanc[n * BN_NP + pp] = acc[r] * inv36;
        }
    }
}

// ---------------- Kernel 3: BatchNorm (training stats) over 4624 values -----
__global__ __launch_bounds__(256) void siam_bn_kernel(
        const float* __restrict__ meanc,
        const float* __restrict__ gamma,
        const float* __restrict__ beta,
        float* __restrict__ out) {
    __shared__ float ssum[256];
    __shared__ float ssq[256];
    const int total = BN_N * BN_NP;          // 4624
    int tid = threadIdx.x;
    float s = 0.f, q = 0.f;
    for (int i = tid; i < total; i += 256) {
        float v = meanc[i];
        s += v; q += v * v;
    }
    ssum[tid] = s; ssq[tid] = q;
    __syncthreads();
    for (int off = 128; off > 0; off >>= 1) {
        if (tid < off) { ssum[tid] += ssum[tid + off]; ssq[tid] += ssq[tid + off]; }
        __syncthreads();
    }
    float inv_n = 1.0f / (float)total;
    float mu  = ssum[0] * inv_n;
    float var = ssq[0] * inv_n - mu * mu;    // biased variance (torch BN)
    float scale = rsqrtf(var + 1e-5f) * gamma[0];
    float shift = beta[0] - mu * scale;
    for (int i = tid; i < total; i += 256) out[i] = meanc[i] * scale + shift;
}

// ---------------- Launch ----------------
extern "C" void kernel_launch(void* const* d_in, const int* in_sizes, int n_in,
                              void* d_out, int out_size, void* d_ws, size_t ws_size,
                              hipStream_t stream) {
    const float* z     = (const float*)d_in[0];   // [16,256,6,6]
    const float* x     = (const float*)d_in[1];   // [16,256,22,22]
    const float* w     = (const float*)d_in[2];   // [1,256,1,1,1]
    const float* gamma = (const float*)d_in[3];   // [1]
    const float* beta  = (const float*)d_in[4];   // [1]

    float* ws    = (float*)d_ws;
    float* sx    = ws;                            // sqrt(x)
    float* szw   = sx + SX_ELEMS;                 // w*sqrt(z)
    float* meanc = szw + SZW_ELEMS;               // [16,289] mean coeff

    int prep_total = SX_ELEMS + SZW_ELEMS;
    siam_prep_kernel<<<(prep_total + 255) / 256, 256, 0, stream>>>(z, x, w, sx, szw);
    siam_corr_kernel<<<BN_N * NTILES, 32, 0, stream>>>(sx, szw, meanc);
    siam_bn_kernel<<<1, 256, 0, stream>>>(meanc, gamma, beta, (float*)d_out);
}